// Attention_84155589198714
// MI455X (gfx1250) — compile-verified
//
#include <hip/hip_runtime.h>
#include <hip/hip_bf16.h>

typedef __attribute__((ext_vector_type(2))) float v2f;
typedef __attribute__((ext_vector_type(8))) float v8f;

#define DIM   1024
#define TD    3072            // 3*DIM
#define NSEQ  2048
#define NB    4
#define NROWS 8192            // NB*NSEQ
#define SROW  2052            // padded LDS row stride (floats): 2052 % 64 == 4 -> conflict-free

__device__ __forceinline__ v8f wmma_f32(v2f a, v2f b, v8f c) {
    // 8 args: (neg_a, A, neg_b, B, c_mod, C, reuse_a, reuse_b)
    return __builtin_amdgcn_wmma_f32_16x16x4_f32(false, a, false, b, (short)0, c, false, false);
}

// ---------------------------------------------------------------------------
// Kernel 1: qkv = context @ W_qkv + b_qkv      [8192,1024]x[1024,3072]
// One wave -> 64x64 macro tile (4x4 WMMA tiles). grid: 128*48/8 = 768 blocks.
// __launch_bounds__(256, 1): allow full VGPR budget so the 128-reg accumulator
// block stays in registers.
// ---------------------------------------------------------------------------
__global__ __launch_bounds__(256, 1)
void qkv_gemm_kernel(const float* __restrict__ ctx,
                     const float* __restrict__ W,
                     const float* __restrict__ bias,
                     float* __restrict__ qkv) {
    const int lane = threadIdx.x & 31;
    const int wv   = threadIdx.x >> 5;
    const int g    = blockIdx.x * 8 + wv;        // 0..6143
    const int gm   = g / 48;                     // 0..127
    const int gn   = g % 48;                     // 0..47
    const int row0 = gm * 64;
    const int col0 = gn * 64;
    const int lm   = lane & 15;                  // M (A) / N (B,C) index
    const int lh   = lane >> 4;                  // lane-half selects K pair / row half

    v8f acc[4][4];
    const v8f vzero = {0.f,0.f,0.f,0.f,0.f,0.f,0.f,0.f};
#pragma unroll
    for (int i = 0; i < 4; ++i)
#pragma unroll
        for (int j = 0; j < 4; ++j) acc[i][j] = vzero;

    for (int k0 = 0; k0 < DIM; k0 += 4) {
        v2f af[4];
#pragma unroll
        for (int i = 0; i < 4; ++i) {
            // A frag: lane lm holds A[row][k0+2*lh], A[row][k0+2*lh+1] (contiguous)
            const float* p = ctx + (size_t)(row0 + 16 * i + lm) * DIM + k0 + 2 * lh;
            af[i] = *(const v2f*)p;
        }
#pragma unroll
        for (int j = 0; j < 4; ++j) {
            // B frag: lane lm holds W[k0+2*lh][col], W[k0+2*lh+1][col] (stride TD)
            const float* bp = W + (size_t)(k0 + 2 * lh) * TD + col0 + 16 * j + lm;
            v2f bf;
            bf.x = bp[0];
            bf.y = bp[TD];
#pragma unroll
            for (int i = 0; i < 4; ++i)
                acc[i][j] = wmma_f32(af[i], bf, acc[i][j]);
        }
    }

    // Epilogue: + bias, store. C/D layout: vgpr v holds row (v + 8*lh), col lm.
#pragma unroll
    for (int j = 0; j < 4; ++j) {
        const int cc = col0 + 16 * j + lm;
        const float bb = bias[cc];
#pragma unroll
        for (int i = 0; i < 4; ++i) {
#pragma unroll
            for (int v = 0; v < 8; ++v) {
                const int rr = row0 + 16 * i + v + 8 * lh;
                qkv[(size_t)rr * TD + cc] = acc[i][j][v] + bb;
            }
        }
    }
}

// ---------------------------------------------------------------------------
// Kernel 2: causal attention for one 16-row Q tile per 128-thread block.
// S row block kept in LDS (16 x 2052 padded fp32). grid: 4*128 = 512 blocks.
// ---------------------------------------------------------------------------
__global__ __launch_bounds__(128, 1)
void attn_kernel(const float* __restrict__ qkv,
                 float* __restrict__ out) {
    extern __shared__ float smem[];
    float* S      = smem;                 // 16 * SROW floats
    float* rowsum = smem + 16 * SROW;     // 16 floats

    const int lane = threadIdx.x & 31;
    const int wv   = threadIdx.x >> 5;    // 0..3
    const int lm   = lane & 15;
    const int lh   = lane >> 4;

    const int it  = blockIdx.x & 127;     // Q row-tile within batch
    const int b   = blockIdx.x >> 7;      // batch
    const int r0  = it * 16;
    const int nct = it + 1;               // causal: col tiles 0..it
    const float scale = 0.022097086912079608f;   // 1/sqrt(2048) (reference quirk)

    const size_t baseRow = (size_t)b * NSEQ;
    const v8f vzero = {0.f,0.f,0.f,0.f,0.f,0.f,0.f,0.f};

    // ---- Pass A: S = scale * Q @ K^T ----
    // Each wave handles tile pairs (2*wv, 2*wv+1), (2*wv+8, 2*wv+9), ...
    // Two independent accumulation chains share each Q A-fragment load.
    const float* qp = qkv + (baseRow + r0 + lm) * TD;          // Q: A-frag row
    for (int jt0 = 2 * wv; jt0 < nct; jt0 += 8) {
        const float* kp0 = qkv + (baseRow + jt0 * 16 + lm) * TD + DIM;
        if (jt0 + 1 < nct) {            // wave-uniform branch (EXEC stays all-1)
            const float* kp1 = kp0 + 16 * TD;
            v8f acc0 = vzero, acc1 = vzero;
            for (int d0 = 0; d0 < DIM; d0 += 4) {
                v2f a  = *(const v2f*)(qp  + d0 + 2 * lh);
                v2f b0 = *(const v2f*)(kp0 + d0 + 2 * lh);
                v2f b1 = *(const v2f*)(kp1 + d0 + 2 * lh);
                acc0 = wmma_f32(a, b0, acc0);
                acc1 = wmma_f32(a, b1, acc1);
            }
#pragma unroll
            for (int i = 0; i < 8; ++i) {
                S[(i + 8 * lh) * SROW + jt0 * 16      + lm] = acc0[i] * scale;
                S[(i + 8 * lh) * SROW + jt0 * 16 + 16 + lm] = acc1[i] * scale;
            }
        } else {
            v8f acc0 = vzero;
            for (int d0 = 0; d0 < DIM; d0 += 4) {
                v2f a  = *(const v2f*)(qp  + d0 + 2 * lh);
                v2f b0 = *(const v2f*)(kp0 + d0 + 2 * lh);
                acc0 = wmma_f32(a, b0, acc0);
            }
#pragma unroll
            for (int i = 0; i < 8; ++i)
                S[(i + 8 * lh) * SROW + jt0 * 16 + lm] = acc0[i] * scale;
        }
    }
    __syncthreads();

    // ---- Softmax in place (8 threads per row); masked tail of diagonal tile -> 0 ----
    {
        const int row   = threadIdx.x >> 3;       // 0..15
        const int sub   = threadIdx.x & 7;
        const int limit = r0 + row + 1;           // valid (causal) columns
        const int total = nct * 16;               // columns written by pass A
        float* Sr = S + row * SROW;

        float mx = -__builtin_inff();
        for (int c = sub; c < limit; c += 8) mx = fmaxf(mx, Sr[c]);
#pragma unroll
        for (int o = 4; o > 0; o >>= 1) mx = fmaxf(mx, __shfl_xor(mx, o, 8));

        float sum = 0.f;
        for (int c = sub; c < total; c += 8) {
            float v = 0.f;
            if (c < limit) { v = __expf(Sr[c] - mx); sum += v; }
            Sr[c] = v;                             // masked/garbage -> exactly 0
        }
#pragma unroll
        for (int o = 4; o > 0; o >>= 1) sum += __shfl_xor(sum, o, 8);
        if (sub == 0) rowsum[row] = sum;
    }
    __syncthreads();

    // ---- Pass B: out = (P @ V) / rowsum ; wave wv owns 256 output dims ----
    const int n0 = wv * 256;
    v8f acc[16];
#pragma unroll
    for (int t = 0; t < 16; ++t) acc[t] = vzero;

    for (int kt = 0; kt < nct; ++kt) {
#pragma unroll
        for (int ks = 0; ks < 4; ++ks) {
            const int k0 = kt * 16 + ks * 4;
            // A frag from LDS: P[lm][k0 + 2*lh], P[lm][k0 + 2*lh + 1]
            v2f a = *(const v2f*)(S + lm * SROW + k0 + 2 * lh);
            // B frags from V: V[key = k0+2*lh(+1)][n0 + 16*nt + lm]
            const float* vp = qkv + (baseRow + k0 + 2 * lh) * TD + 2 * DIM + n0 + lm;
#pragma unroll
            for (int nt = 0; nt < 16; ++nt) {
                v2f bb;
                bb.x = vp[nt * 16];
                bb.y = vp[nt * 16 + TD];
                acc[nt] = wmma_f32(a, bb, acc[nt]);
            }
        }
    }

#pragma unroll
    for (int nt = 0; nt < 16; ++nt) {
#pragma unroll
        for (int i = 0; i < 8; ++i) {
            const int m = i + 8 * lh;
            const float val = acc[nt][i] * (1.0f / rowsum[m]);
            out[(baseRow + r0 + m) * DIM + n0 + nt * 16 + lm] = val;
        }
    }
}

// ---------------------------------------------------------------------------
extern "C" void kernel_launch(void* const* d_in, const int* in_sizes, int n_in,
                              void* d_out, int out_size, void* d_ws, size_t ws_size,
                              hipStream_t stream) {
    const float* ctx  = (const float*)d_in[0];   // [4,2048,1024]
    const float* W    = (const float*)d_in[1];   // [1024,3072]
    const float* bias = (const float*)d_in[2];   // [3072]
    float* out = (float*)d_out;                  // [4,2048,1024]
    float* qkv = (float*)d_ws;                   // [8192,3072] fp32 = 96 MB scratch

    // Kernel 1: 128 x 48 macro-tiles of 64x64, 8 waves (256 thr) per block.
    qkv_gemm_kernel<<<768, 256, 0, stream>>>(ctx, W, bias, qkv);

    // Kernel 2: one block per (batch, 16-row Q tile).
    const size_t lds_bytes = (size_t)(16 * SROW + 16) * sizeof(float);
    attn_kernel<<<NB * (NSEQ / 16), 128, lds_bytes, stream>>>(qkv, out);
}